// ICP_25623774888437
// MI455X (gfx1250) — compile-verified
//
#include <hip/hip_runtime.h>
#include <math.h>

typedef __attribute__((ext_vector_type(2))) float v2f;
typedef __attribute__((ext_vector_type(4))) float v4f;
typedef __attribute__((ext_vector_type(8))) float v8f;

#define ICP_TOL   1e-4f
#define ICP_STEPS 20   // reference runs STEPLIM+1 = 21 masked iterations
#define NSEG      8    // column-range split for occupancy

// ---------------------------------------------------------------------------
// init: pc = p1; p2p[i] = {x, y, z, ||p||^2}; state = {err=0, done=0}
// ---------------------------------------------------------------------------
__global__ void icp_init(const float* __restrict__ p1, const float* __restrict__ p2,
                         float* __restrict__ pc, v4f* __restrict__ p2p,
                         float* __restrict__ state, int N, int M) {
  int i = blockIdx.x * blockDim.x + threadIdx.x;
  if (i < N) {
    pc[3*i+0] = p1[3*i+0];
    pc[3*i+1] = p1[3*i+1];
    pc[3*i+2] = p1[3*i+2];
  }
  if (i < M) {
    float x = p2[3*i+0], y = p2[3*i+1], z = p2[3*i+2];
    v4f q; q.x = x; q.y = y; q.z = z; q.w = x*x + y*y + z*z;
    p2p[i] = q;
  }
  if (i == 0) { state[0] = 0.0f; state[1] = 0.0f; }
}

// ---------------------------------------------------------------------------
// 1-NN via V_WMMA_F32_16X16X4_F32, norms folded into the MMA:
//   A row m  = (-2x, -2y, -2z, 1)            (pc)
//   B col n  = ( x,   y,   z,  ||p2_n||^2)   (p2 padded float4)
//   C[m][n]  = ||pc_m||^2                    (loop-invariant!)
//   => D = d^2 directly; zero per-tile C rebuild.
// B operand lands via one global_load_b64 per lane:
//   lanes 0-15 need (x,y) pair, lanes 16-31 need (z,n2) pair of the same col.
// A layout: lanes 0-15 row=lane K={0,1}; lanes 16-31 row=lane-16 K={2,3}.
// C/D layout: VGPR v -> row (lane<16 ? v : v+8), col = lane&15.
// OOB columns are clamped duplicates of col M-1; strict-< and index tie-break
// guarantee they never displace the genuine argmin.
// ---------------------------------------------------------------------------
__global__ __launch_bounds__(128)
void icp_nn_wmma(const float* __restrict__ pc, const v2f* __restrict__ p2pairs,
                 float* __restrict__ pdmin, int* __restrict__ pidx,
                 int N, int M) {
  const int lane = threadIdx.x & 31;
  const int w    = threadIdx.x >> 5;
  const int seg  = blockIdx.y;
  const int nRowTiles = (N + 15) >> 4;
  const int rtileReal = blockIdx.x * 4 + w;
  const int rtile = (rtileReal < nRowTiles) ? rtileReal : (nRowTiles - 1);

  __shared__ float n1s[4][16];

  const bool hi    = (lane >= 16);
  const int  rslot = lane & 15;

  int r = rtile * 16 + rslot; if (r >= N) r = N - 1;
  const float x = pc[3*r+0], y = pc[3*r+1], z = pc[3*r+2];

  v2f a;
  a.x = hi ? (-2.0f * z) : (-2.0f * x);
  a.y = hi ? 1.0f        : (-2.0f * y);
  if (!hi) n1s[w][rslot] = x*x + y*y + z*z;
  __syncthreads();

  const int rowbase = hi ? 8 : 0;
  v8f c;  // loop-invariant accumulator input: C[m][n] = ||pc_m||^2
#pragma unroll
  for (int v = 0; v < 8; ++v) c[v] = n1s[w][rowbase + v];

  float best[8]; int bcol[8];
#pragma unroll
  for (int v = 0; v < 8; ++v) { best[v] = 3.0e38f; bcol[v] = 0; }

  const int nColTiles = (M + 15) >> 4;
  const int tilesPerSeg = (nColTiles + NSEG - 1) / NSEG;
  const int ctBeg = seg * tilesPerSeg;
  int ctEnd = ctBeg + tilesPerSeg; if (ctEnd > nColTiles) ctEnd = nColTiles;

  const int hioff = hi ? 1 : 0;

#pragma unroll 2
  for (int ct = ctBeg; ct < ctEnd; ++ct) {
    const int cidx = ct * 16 + rslot;
    const int cld  = (cidx < M) ? cidx : (M - 1);
    // single global_load_b64: B operand directly in place, no fixup VALU
    const v2f b = p2pairs[cld * 2 + hioff];
    v8f d = __builtin_amdgcn_wmma_f32_16x16x4_f32(
        /*neg_a=*/false, a, /*neg_b=*/false, b,
        /*c_mod=*/(short)0, c, /*reuse_a=*/false, /*reuse_b=*/false);
#pragma unroll
    for (int v = 0; v < 8; ++v)
      if (d[v] < best[v]) { best[v] = d[v]; bcol[v] = cidx; }
  }

  // reduce (min, argmin) across the 16 lanes of each half-wave
#pragma unroll
  for (int v = 0; v < 8; ++v) {
    float bd = best[v]; int bc = bcol[v];
#pragma unroll
    for (int s = 1; s < 16; s <<= 1) {
      float od = __shfl_xor(bd, s, 32);
      int   oc = __shfl_xor(bc, s, 32);
      if (od < bd || (od == bd && oc < bc)) { bd = od; bc = oc; }
    }
    const int rr = rtile * 16 + rowbase + v;
    if (rslot == 0 && rr < N && rtileReal < nRowTiles) {
      pdmin[seg * N + rr] = bd;   // keep d^2; sqrt in the reduce pass
      pidx [seg * N + rr] = bc;
    }
  }
}

// ---------------------------------------------------------------------------
// combine per-segment partial argmins (first-min-index tie break)
// ---------------------------------------------------------------------------
__global__ void icp_nn_reduce(const float* __restrict__ pdmin, const int* __restrict__ pidx,
                              float* __restrict__ dmin, int* __restrict__ nnidx, int N) {
  const int r = blockIdx.x * blockDim.x + threadIdx.x;
  if (r >= N) return;
  float bd = 3.0e38f; int bc = 0x7fffffff;
#pragma unroll
  for (int s = 0; s < NSEG; ++s) {
    const float d = pdmin[s * N + r];
    const int   i = pidx [s * N + r];
    if (d < bd || (d == bd && i < bc)) { bd = d; bc = i; }
  }
  dmin[r]  = sqrtf(fmaxf(bd, 1e-12f));
  nnidx[r] = bc;
}

// ---------------------------------------------------------------------------
// Kabsch + apply (single block). mode 0: ICP step (dst = p2[idx], update pc,
// state, err). mode 1: final (dst = pc identity-paired, write [R|t] to out).
// ---------------------------------------------------------------------------
__device__ inline float det3(const float m[3][3]) {
  return m[0][0]*(m[1][1]*m[2][2]-m[1][2]*m[2][1])
       - m[0][1]*(m[1][0]*m[2][2]-m[1][2]*m[2][0])
       + m[0][2]*(m[1][0]*m[2][1]-m[1][1]*m[2][0]);
}

__global__ __launch_bounds__(256)
void icp_kabsch(const float* __restrict__ src, const float* __restrict__ dst,
                const int* __restrict__ nnidx, const float* __restrict__ dmin,
                float* __restrict__ pc, float* __restrict__ state,
                float* __restrict__ out, int N, int mode) {
  const int tid = threadIdx.x;
  float acc[16];
#pragma unroll
  for (int q = 0; q < 16; ++q) acc[q] = 0.0f;

  for (int n = tid; n < N; n += 256) {
    const float sx = src[3*n+0], sy = src[3*n+1], sz = src[3*n+2];
    const int m = (mode == 0) ? nnidx[n] : n;
    const float dx = dst[3*m+0], dy = dst[3*m+1], dz = dst[3*m+2];
    acc[0] += sx; acc[1] += sy; acc[2] += sz;
    acc[3] += dx; acc[4] += dy; acc[5] += dz;
    acc[6]  += sx*dx; acc[7]  += sx*dy; acc[8]  += sx*dz;
    acc[9]  += sy*dx; acc[10] += sy*dy; acc[11] += sy*dz;
    acc[12] += sz*dx; acc[13] += sz*dy; acc[14] += sz*dz;
    if (mode == 0) acc[15] += dmin[n];
  }

  __shared__ float red[16 * 256];
#pragma unroll
  for (int q = 0; q < 16; ++q) red[q * 256 + tid] = acc[q];
  __syncthreads();
  for (int off = 128; off > 0; off >>= 1) {
    if (tid < off) {
#pragma unroll
      for (int q = 0; q < 16; ++q) red[q*256 + tid] += red[q*256 + tid + off];
    }
    __syncthreads();
  }

  __shared__ float sRt[13]; // R[9], t[3], apply flag
  if (tid == 0) {
    const float invN = 1.0f / (float)N;
    const float c1[3] = { red[0*256]*invN, red[1*256]*invN, red[2*256]*invN };
    const float c2[3] = { red[3*256]*invN, red[4*256]*invN, red[5*256]*invN };
    float H[3][3];
    for (int i = 0; i < 3; ++i)
      for (int j = 0; j < 3; ++j)
        H[i][j] = red[(6 + i*3 + j) * 256] - (float)N * c1[i] * c2[j];

    // Jacobi eigendecomposition of A = H^T H -> V (right singular vectors)
    float A[3][3], V[3][3];
    for (int i = 0; i < 3; ++i)
      for (int j = 0; j < 3; ++j) {
        float s = 0.0f;
        for (int k = 0; k < 3; ++k) s += H[k][i] * H[k][j];
        A[i][j] = s;
        V[i][j] = (i == j) ? 1.0f : 0.0f;
      }
    for (int sweep = 0; sweep < 12; ++sweep) {
      for (int pq = 0; pq < 3; ++pq) {
        const int p = (pq == 2) ? 1 : 0;
        const int q = (pq == 0) ? 1 : 2;
        const float apq = A[p][q];
        if (fabsf(apq) < 1e-20f) continue;
        const float theta = (A[q][q] - A[p][p]) / (2.0f * apq);
        const float tt = ((theta >= 0.0f) ? 1.0f : -1.0f) /
                         (fabsf(theta) + sqrtf(1.0f + theta*theta));
        const float cc = 1.0f / sqrtf(1.0f + tt*tt);
        const float ss = tt * cc;
        for (int k = 0; k < 3; ++k) { float kp=A[k][p], kq=A[k][q]; A[k][p]=cc*kp-ss*kq; A[k][q]=ss*kp+cc*kq; }
        for (int k = 0; k < 3; ++k) { float pk=A[p][k], qk=A[q][k]; A[p][k]=cc*pk-ss*qk; A[q][k]=ss*pk+cc*qk; }
        for (int k = 0; k < 3; ++k) { float kp=V[k][p], kq=V[k][q]; V[k][p]=cc*kp-ss*kq; V[k][q]=ss*kp+cc*kq; }
      }
    }
    // sort eigenvalues descending, permute V columns
    float ev[3] = { A[0][0], A[1][1], A[2][2] };
    int ord[3] = {0, 1, 2};
    for (int i = 0; i < 2; ++i)
      for (int j = i + 1; j < 3; ++j)
        if (ev[ord[j]] > ev[ord[i]]) { int t0 = ord[i]; ord[i] = ord[j]; ord[j] = t0; }
    float Vs[3][3];
    for (int j = 0; j < 3; ++j)
      for (int i = 0; i < 3; ++i) Vs[i][j] = V[i][ord[j]];

    // U columns: u_j = H v_j / s_j (normalize); fix rank-deficient last column
    float U[3][3];
    for (int j = 0; j < 3; ++j) {
      float u0 = H[0][0]*Vs[0][j] + H[0][1]*Vs[1][j] + H[0][2]*Vs[2][j];
      float u1 = H[1][0]*Vs[0][j] + H[1][1]*Vs[1][j] + H[1][2]*Vs[2][j];
      float u2 = H[2][0]*Vs[0][j] + H[2][1]*Vs[1][j] + H[2][2]*Vs[2][j];
      float len = sqrtf(u0*u0 + u1*u1 + u2*u2);
      if (len > 1e-20f) { U[0][j]=u0/len; U[1][j]=u1/len; U[2][j]=u2/len; }
      else if (j == 2) {
        float cx = U[1][0]*U[2][1] - U[2][0]*U[1][1];
        float cy = U[2][0]*U[0][1] - U[0][0]*U[2][1];
        float cz = U[0][0]*U[1][1] - U[1][0]*U[0][1];
        float cl = sqrtf(cx*cx + cy*cy + cz*cz) + 1e-30f;
        U[0][2]=cx/cl; U[1][2]=cy/cl; U[2][2]=cz/cl;
      } else { U[0][j]=(j==0); U[1][j]=(j==1); U[2][j]=0.0f; }
    }

    const float dd = (det3(U) * det3(Vs) >= 0.0f) ? 1.0f : -1.0f;
    const float D3[3] = { 1.0f, 1.0f, dd };
    float R[3][3];
    for (int i = 0; i < 3; ++i)
      for (int j = 0; j < 3; ++j) {
        float s = 0.0f;
        for (int k = 0; k < 3; ++k) s += Vs[i][k] * D3[k] * U[j][k];
        R[i][j] = s;
      }
    float t[3];
    for (int i = 0; i < 3; ++i)
      t[i] = c2[i] - (R[i][0]*c1[0] + R[i][1]*c1[1] + R[i][2]*c1[2]);

    if (mode == 0) {
      const float errnew  = red[15*256];   // /B with B==1
      const float errold  = state[0];
      const float doneold = state[1];
      const float apply   = (doneold == 0.0f) ? 1.0f : 0.0f;
      if (doneold == 0.0f) {
        state[0] = errnew;
        if (fabsf(errold - errnew) < ICP_TOL) state[1] = 1.0f;
      }
      for (int k = 0; k < 9; ++k) sRt[k] = R[k/3][k%3];
      sRt[9] = t[0]; sRt[10] = t[1]; sRt[11] = t[2];
      sRt[12] = apply;
    } else {
      out[0]=R[0][0]; out[1]=R[0][1]; out[2] =R[0][2]; out[3] =t[0];
      out[4]=R[1][0]; out[5]=R[1][1]; out[6] =R[1][2]; out[7] =t[1];
      out[8]=R[2][0]; out[9]=R[2][1]; out[10]=R[2][2]; out[11]=t[2];
      sRt[12] = 0.0f;
    }
  }
  __syncthreads();

  if (mode == 0 && sRt[12] != 0.0f) {
    const float R00=sRt[0], R01=sRt[1], R02=sRt[2];
    const float R10=sRt[3], R11=sRt[4], R12=sRt[5];
    const float R20=sRt[6], R21=sRt[7], R22=sRt[8];
    const float t0=sRt[9], t1=sRt[10], t2=sRt[11];
    for (int n = tid; n < N; n += 256) {
      const float px = pc[3*n+0], py = pc[3*n+1], pz = pc[3*n+2];
      pc[3*n+0] = R00*px + R01*py + R02*pz + t0;
      pc[3*n+1] = R10*px + R11*py + R12*pz + t1;
      pc[3*n+2] = R20*px + R21*py + R22*pz + t2;
    }
  }
}

// ---------------------------------------------------------------------------
extern "C" void kernel_launch(void* const* d_in, const int* in_sizes, int n_in,
                              void* d_out, int out_size, void* d_ws, size_t ws_size,
                              hipStream_t stream) {
  const float* p1 = (const float*)d_in[0];
  const float* p2 = (const float*)d_in[1];
  const int N = in_sizes[0] / 3;
  const int M = in_sizes[1] / 3;

  float* ws    = (float*)d_ws;
  float* pc    = ws;                     // 3N floats: current transformed p1
  v4f*   p2p   = (v4f*)(ws + 3*N);       // 4M floats: {x,y,z,||p||^2} (16B aligned)
  float* dmin  = ws + 3*N + 4*M;         // N floats
  int*   nnidx = (int*)(ws + 4*N + 4*M); // N ints
  float* pdmin = ws + 5*N + 4*M;         // NSEG*N floats: per-segment partial d^2
  int*   pidx  = (int*)(ws + 5*N + 4*M + NSEG*N); // NSEG*N ints
  float* state = ws + 5*N + 4*M + 2*NSEG*N;       // {err, done}
  float* out   = (float*)d_out;          // 12 floats: [R|t]

  const int mx = (N > M) ? N : M;
  icp_init<<<(mx + 255) / 256, 256, 0, stream>>>(p1, p2, pc, p2p, state, N, M);

  const int nRowTiles = (N + 15) >> 4;
  dim3 nnGrid((nRowTiles + 3) / 4, NSEG);
  for (int s = 0; s <= ICP_STEPS; ++s) {
    icp_nn_wmma<<<nnGrid, 128, 0, stream>>>(pc, (const v2f*)p2p, pdmin, pidx, N, M);
    icp_nn_reduce<<<(N + 255) / 256, 256, 0, stream>>>(pdmin, pidx, dmin, nnidx, N);
    icp_kabsch<<<1, 256, 0, stream>>>(pc, p2, nnidx, dmin, pc, state, nullptr, N, 0);
  }
  // final rigid transform between original p1 and converged pc
  icp_kabsch<<<1, 256, 0, stream>>>(p1, pc, nullptr, nullptr, pc, state, out, N, 1);
}